// GCN_12799002542568
// MI455X (gfx1250) — compile-verified
//
#include <hip/hip_runtime.h>
#include <hip/hip_bf16.h>

typedef __attribute__((ext_vector_type(2))) float v2f;
typedef __attribute__((ext_vector_type(8))) float v8f;

// ---------------------------------------------------------------------------
// Pass 1: init. deg starts at 1.0 (self loop); zero the accumulators.
// ---------------------------------------------------------------------------
__global__ void gcn_init(float* __restrict__ deg, float* __restrict__ t,
                         float* __restrict__ agg2, int n) {
  int i = blockIdx.x * blockDim.x + threadIdx.x;
  if (i < n) {
    deg[i] = 1.0f;
    t[i] = 0.0f;
    agg2[2 * i + 0] = 0.0f;
    agg2[2 * i + 1] = 0.0f;
  }
}

// ---------------------------------------------------------------------------
// Pass 2: in-degree scatter (counts of dst).
// ---------------------------------------------------------------------------
__global__ void gcn_deg(const int* __restrict__ dst, float* __restrict__ deg,
                        int e) {
  int i = blockIdx.x * blockDim.x + threadIdx.x;
  if (i < e) atomicAdd(&deg[dst[i]], 1.0f);
}

// ---------------------------------------------------------------------------
// Pass 3: dis = rsqrt(deg) (deg >= 1 always), v = dis * x.
// ---------------------------------------------------------------------------
__global__ void gcn_disv(const float* __restrict__ x, float* __restrict__ dis,
                         float* __restrict__ v, int n) {
  int i = blockIdx.x * blockDim.x + threadIdx.x;
  if (i < n) {
    float r = rsqrtf(dis[i]);  // dis currently holds deg
    dis[i] = r;
    v[i] = r * x[i];
  }
}

// ---------------------------------------------------------------------------
// Pass 4: layer-1 scalar scatter: t[dst] += v[src].
// ---------------------------------------------------------------------------
__global__ void gcn_scat1(const int* __restrict__ src,
                          const int* __restrict__ dst,
                          const float* __restrict__ v, float* __restrict__ t,
                          int e) {
  int i = blockIdx.x * blockDim.x + threadIdx.x;
  if (i < e) atomicAdd(&t[dst[i]], v[src[i]]);
}

// ---------------------------------------------------------------------------
// Pass 5: per-node transform via WMMA.
//   s   = dis * (t + v)                         (layer-1 aggregated scalar)
//   h1j = relu(s * W1[j] + b1[j])   j=0..15
//   z   = h1 @ W2  (16x2)  -> 16(node)x16 WMMA tile, cols 2..15 are zero
//   zz  = dis * z   (pre-scaled by dis[src] for the layer-2 scatter)
//
// One wave handles 16 nodes. V_WMMA_F32_16X16X4_F32 chained 4x for K=16.
// A layout (ISA 7.12.2, 32-bit A 16x4): lanes 0-15 hold row M=lane with
// K={0,1} in v0/v1; lanes 16-31 hold the same rows with K={2,3}.
// B (4x16) row-striped across lanes within a VGPR: v0 = rows {K0 | K2},
// v1 = rows {K1 | K3} (mirrors A / C conventions).
// C/D layout: vgpr r, lanes 0-15 -> D[r][lane]; lanes 16-31 -> D[8+r][lane-16].
//
// Each lane computes only the 8 h-values it feeds into A (runtime-addressed
// loads of W1/b1 -- no dynamic register indexing), and W2 is loaded at a
// clamped in-bounds index with a value select (no exec-masked branches).
// ---------------------------------------------------------------------------
__global__ void gcn_node_wmma(const float* __restrict__ dis,
                              const float* __restrict__ t,
                              const float* __restrict__ v,
                              const float* __restrict__ W1,
                              const float* __restrict__ b1,
                              const float* __restrict__ W2,
                              float* __restrict__ zz, int n) {
  const int lane = threadIdx.x & 31;
  const int wave = blockIdx.x * (blockDim.x >> 5) + (threadIdx.x >> 5);
  const int base = wave * 16;
  if (base >= n) return;  // wave-uniform: EXEC stays all-ones for WMMA

  const int row = lane & 15;
  const bool hi = lane >= 16;
  const int off = hi ? 2 : 0;          // this lane's K sub-pair
  const int rc = (row < 2) ? row : 0;  // clamped W2 column (in-bounds)
  const float bsel = (row < 2) ? 1.0f : 0.0f;
  const int node = base + row;

  const float d = dis[node];
  const float s = d * (t[node] + v[node]);

  v8f c = {0.f, 0.f, 0.f, 0.f, 0.f, 0.f, 0.f, 0.f};
#pragma unroll
  for (int q = 0; q < 4; ++q) {
    const int j0 = 4 * q + off;  // feature index for this lane's K slots
    v2f a;
    a.x = fmaxf(fmaf(s, W1[j0 + 0], b1[j0 + 0]), 0.0f);
    a.y = fmaxf(fmaf(s, W1[j0 + 1], b1[j0 + 1]), 0.0f);
    v2f b;
    b.x = bsel * W2[(j0 + 0) * 2 + rc];  // W2 is [16,2] row-major
    b.y = bsel * W2[(j0 + 1) * 2 + rc];
    c = __builtin_amdgcn_wmma_f32_16x16x4_f32(
        /*neg_a=*/false, a, /*neg_b=*/false, b,
        /*c_mod=*/(short)0, c, /*reuse_a=*/false, /*reuse_b=*/false);
  }

  // Only output columns 0 and 1 are live: lanes 0,1 carry rows 0..7,
  // lanes 16,17 carry rows 8..15.
  if ((lane & 15) < 2) {
    const int col = lane & 1;
    const int rb = hi ? 8 : 0;
#pragma unroll
    for (int r = 0; r < 8; ++r) {
      const int nd = base + rb + r;
      zz[2 * nd + col] = dis[nd] * c[r];
    }
  }
}

// ---------------------------------------------------------------------------
// Pass 6: layer-2 scatter: agg2[dst] += zz[src]  (2 floats/edge).
// ---------------------------------------------------------------------------
__global__ void gcn_scat2(const int* __restrict__ src,
                          const int* __restrict__ dst,
                          const float2* __restrict__ zz2,
                          float* __restrict__ agg2, int e) {
  int i = blockIdx.x * blockDim.x + threadIdx.x;
  if (i < e) {
    const int sN = src[i];
    const int dN = dst[i];
    const float2 z = zz2[sN];
    atomicAdd(&agg2[2 * dN + 0], z.x);
    atomicAdd(&agg2[2 * dN + 1], z.y);
  }
}

// ---------------------------------------------------------------------------
// Pass 7: finalize: p = dis*(agg2 + zz) + b2, then stable 2-way log_softmax.
// ---------------------------------------------------------------------------
__global__ void gcn_final(const float* __restrict__ dis,
                          const float* __restrict__ zz,
                          const float* __restrict__ agg2,
                          const float* __restrict__ b2,
                          float* __restrict__ out, int n) {
  int i = blockIdx.x * blockDim.x + threadIdx.x;
  if (i < n) {
    const float d = dis[i];
    const float p0 = fmaf(d, agg2[2 * i + 0] + zz[2 * i + 0], b2[0]);
    const float p1 = fmaf(d, agg2[2 * i + 1] + zz[2 * i + 1], b2[1]);
    const float m = fmaxf(p0, p1);
    const float lse = m + log1pf(expf(-fabsf(p0 - p1)));
    out[2 * i + 0] = p0 - lse;
    out[2 * i + 1] = p1 - lse;
  }
}

extern "C" void kernel_launch(void* const* d_in, const int* in_sizes, int n_in,
                              void* d_out, int out_size, void* d_ws,
                              size_t ws_size, hipStream_t stream) {
  (void)n_in;
  (void)out_size;
  (void)ws_size;

  const float* x = (const float*)d_in[0];
  const int* ei = (const int*)d_in[1];  // int32 per harness contract
  const float* W1 = (const float*)d_in[2];
  const float* b1 = (const float*)d_in[3];
  const float* W2 = (const float*)d_in[4];
  const float* b2 = (const float*)d_in[5];
  float* out = (float*)d_out;

  const int n = in_sizes[0];      // 100000 (x is [N,1])
  const int e = in_sizes[1] / 2;  // 3200000
  const int* src = ei;            // edge_index row 0
  const int* dst = ei + e;        // edge_index row 1

  // Workspace layout (floats): dis[n] | v[n] | t[n] | zz[2n] | agg2[2n]
  float* ws = (float*)d_ws;
  float* dis = ws;
  float* v = ws + (size_t)n;
  float* t = ws + (size_t)2 * n;
  float* zz = ws + (size_t)3 * n;
  float* agg2 = ws + (size_t)5 * n;

  const int TB = 256;
  const int gN = (n + TB - 1) / TB;
  const int gE = (e + TB - 1) / TB;
  const int gW = (n + 8 * 16 - 1) / (8 * 16);  // 8 waves/block, 16 nodes/wave

  gcn_init<<<gN, TB, 0, stream>>>(dis, t, agg2, n);
  gcn_deg<<<gE, TB, 0, stream>>>(dst, dis, e);
  gcn_disv<<<gN, TB, 0, stream>>>(x, dis, v, n);
  gcn_scat1<<<gE, TB, 0, stream>>>(src, dst, v, t, e);
  gcn_node_wmma<<<gW, TB, 0, stream>>>(dis, t, v, W1, b1, W2, zz, n);
  gcn_scat2<<<gE, TB, 0, stream>>>(src, dst, (const float2*)zz, agg2, e);
  gcn_final<<<gN, TB, 0, stream>>>(dis, zz, agg2, b2, out, n);
}